// _VarGate_65481071399964
// MI455X (gfx1250) — compile-verified
//
#include <hip/hip_runtime.h>
#include <hip/hip_bf16.h>
#include <math.h>

#define N_WIRES 12
#define DEPTH   4
#define IN_DIM  256
#define BATCH   4096
#define NSTATES 4096   // 2^12

typedef __attribute__((ext_vector_type(2))) float v2f;
typedef __attribute__((ext_vector_type(8))) float v8f;

// ---------------------------------------------------------------------------
// Kernel A: angles[b][w] = sum_k x[b][k] * W[w][k]   (padded to 16 columns)
// One wave32 computes a 16x16 tile with V_WMMA_F32_16X16X4_F32, K=256 in 64
// steps. A layout: lane(l&15)=M, VGPR j holds K = 2*(l>>4)+j. B mirrored.
// Padding rows of B are clamped (not zeroed): GEMM columns are independent,
// and output columns 12-15 are never read, so no predication is needed.
// ---------------------------------------------------------------------------
__global__ __launch_bounds__(256) void vq_angles_wmma(
    const float* __restrict__ x, const float* __restrict__ W,
    float* __restrict__ angles /* BATCH x 16 */)
{
  const int lane = threadIdx.x & 31;
  const int wave = threadIdx.x >> 5;
  const int row0 = (blockIdx.x * 8 + wave) * 16;   // 16-row tile
  const int half = lane >> 4;                      // K sub-pair select
  const int mrow = lane & 15;                      // A:M index / B:N index

  const int wrow = (mrow < N_WIRES) ? mrow : 0;    // clamp padding rows
  const float* xr = x + (size_t)(row0 + mrow) * IN_DIM;
  const float* wr = W + (size_t)wrow * IN_DIM;

  v8f acc = {0.f, 0.f, 0.f, 0.f, 0.f, 0.f, 0.f, 0.f};
  #pragma unroll 8
  for (int k0 = 0; k0 < IN_DIM; k0 += 4) {
    const int kk = k0 + 2 * half;
    v2f a, b;
    a.x = xr[kk];
    a.y = xr[kk + 1];
    b.x = wr[kk];
    b.y = wr[kk + 1];
    acc = __builtin_amdgcn_wmma_f32_16x16x4_f32(false, a, false, b,
                                                (short)0, acc, false, false);
  }
  // C/D: VGPR r -> lanes 0-15: (M=r, N=lane); lanes 16-31: (M=r+8, N=lane-16)
  const int ncol = lane & 15;
  #pragma unroll
  for (int r = 0; r < 8; ++r) {
    const int m = r + 8 * half;
    angles[(size_t)(row0 + m) * 16 + ncol] = acc[r];
  }
}

// ---------------------------------------------------------------------------
// Kernel B: full 12-qubit statevector simulation, one workgroup per batch
// element, state (4096 complex) resident in LDS (32 KB of 320 KB/WGP).
//  - encoding RX layer folded into closed-form product state
//  - each CNOT ladder collapsed to Gray-code permutation new[i]=old[i^(i>>1)]
//  - <Z_w> via in-LDS tree reduction
// ---------------------------------------------------------------------------
__global__ __launch_bounds__(256) void vq_sim(
    const float* __restrict__ angles /* BATCH x 16 */,
    const float* __restrict__ params /* DEPTH*N_WIRES */,
    float* __restrict__ out /* BATCH x N_WIRES */)
{
  __shared__ float sre[NSTATES];
  __shared__ float sim_[NSTATES];
  __shared__ float sc[N_WIRES], ss[N_WIRES];
  __shared__ float cp[DEPTH * N_WIRES], sp[DEPTH * N_WIRES];

  const int tid = threadIdx.x;
  const int b   = blockIdx.x;

  if (tid < N_WIRES) {
    float h = 0.5f * angles[(size_t)b * 16 + tid];
    sc[tid] = cosf(h);
    ss[tid] = sinf(h);
  }
  if (tid < DEPTH * N_WIRES) {
    float h = 0.5f * params[tid];
    cp[tid] = cosf(h);
    sp[tid] = sinf(h);
  }
  __syncthreads();

  // Data-dependent trig into registers.
  float lc[N_WIRES], ls[N_WIRES];
  #pragma unroll
  for (int w = 0; w < N_WIRES; ++w) { lc[w] = sc[w]; ls[w] = ss[w]; }

  // Product state after encoding layer: amp(i) = (prod mags) * (-i)^popc(i).
  // Wire w maps to bit (11-w).
  #pragma unroll
  for (int k = 0; k < 16; ++k) {
    int idx = tid * 16 + k;
    float m = 1.f;
    #pragma unroll
    for (int p = 0; p < N_WIRES; ++p)
      m *= ((idx >> p) & 1) ? ls[11 - p] : lc[11 - p];
    int kp = __popc(idx) & 3;
    float re = 0.f, im = 0.f;
    if      (kp == 0) re =  m;
    else if (kp == 1) im = -m;
    else if (kp == 2) re = -m;
    else              im =  m;
    sre[idx]  = re;
    sim_[idx] = im;
  }

  for (int d = 0; d < DEPTH; ++d) {
    // 12 parameterized RX sweeps over the LDS state.
    #pragma unroll 1
    for (int w = 0; w < N_WIRES; ++w) {
      __syncthreads();
      const float c = cp[d * N_WIRES + w];
      const float s = sp[d * N_WIRES + w];
      const int S = 1 << (11 - w);
      #pragma unroll
      for (int r = 0; r < 8; ++r) {
        int p  = r * 256 + tid;                       // pair id 0..2047
        int i0 = ((p & ~(S - 1)) << 1) | (p & (S - 1));
        int i1 = i0 + S;
        float a0r = sre[i0], a0i = sim_[i0];
        float a1r = sre[i1], a1i = sim_[i1];
        // b0 = c*a0 - i*s*a1 ; b1 = -i*s*a0 + c*a1
        sre[i0]  = fmaf(c, a0r,  s * a1i);
        sim_[i0] = fmaf(c, a0i, -s * a1r);
        sre[i1]  = fmaf(c, a1r,  s * a0i);
        sim_[i1] = fmaf(c, a1i, -s * a0r);
      }
    }
    __syncthreads();
    // CNOT ladder (ctrl w -> w+1, w=0..10) == Gray-code gather.
    float rr[16], ri[16];
    #pragma unroll
    for (int k = 0; k < 16; ++k) {
      int i = tid * 16 + k;
      int j = i ^ (i >> 1);
      rr[k] = sre[j];
      ri[k] = sim_[j];
    }
    __syncthreads();
    #pragma unroll
    for (int k = 0; k < 16; ++k) {
      int i = tid * 16 + k;
      sre[i]  = rr[k];
      sim_[i] = ri[k];
    }
  }
  __syncthreads();

  // <Z_w> partial sums (sign = 1-2*bit_{11-w}).
  float part[N_WIRES];
  #pragma unroll
  for (int w = 0; w < N_WIRES; ++w) part[w] = 0.f;
  #pragma unroll
  for (int k = 0; k < 16; ++k) {
    int idx = tid * 16 + k;
    float pr = sre[idx] * sre[idx] + sim_[idx] * sim_[idx];
    #pragma unroll
    for (int w = 0; w < N_WIRES; ++w)
      part[w] += ((idx >> (11 - w)) & 1) ? -pr : pr;
  }
  __syncthreads();
  #pragma unroll
  for (int w = 0; w < N_WIRES; ++w) sre[tid * N_WIRES + w] = part[w];
  __syncthreads();
  for (int off = 128; off >= 1; off >>= 1) {
    if (tid < off) {
      #pragma unroll
      for (int w = 0; w < N_WIRES; ++w)
        sre[tid * N_WIRES + w] += sre[(tid + off) * N_WIRES + w];
    }
    __syncthreads();
  }
  if (tid < N_WIRES) out[(size_t)b * N_WIRES + tid] = sre[tid];
}

// ---------------------------------------------------------------------------
extern "C" void kernel_launch(void* const* d_in, const int* in_sizes, int n_in,
                              void* d_out, int out_size, void* d_ws, size_t ws_size,
                              hipStream_t stream) {
  (void)in_sizes; (void)n_in; (void)out_size; (void)ws_size;
  const float* x      = (const float*)d_in[0];   // (4096, 256)
  const float* W      = (const float*)d_in[1];   // (12, 256)
  const float* params = (const float*)d_in[2];   // (4, 12, 1)
  float* out    = (float*)d_out;                 // (4096, 12)
  float* angles = (float*)d_ws;                  // (4096, 16) fp32 = 256 KB

  // 8 waves/block, one 16x16 WMMA tile per wave -> 4096/128 = 32 blocks.
  vq_angles_wmma<<<BATCH / 128, 256, 0, stream>>>(x, W, angles);
  // One workgroup per batch element, statevector in LDS.
  vq_sim<<<BATCH, 256, 0, stream>>>(angles, params, out);
}